// AverageMeshNetworkPEARAR_86303072845950
// MI455X (gfx1250) — compile-verified
//
#include <hip/hip_runtime.h>

typedef __attribute__((ext_vector_type(2))) float v2f;
typedef __attribute__((ext_vector_type(8))) float v8f;

#define EPS_N  1e-5f
#define LSLOPE 0.01f

__device__ __forceinline__ float lrelu(float x) { return x >= 0.f ? x : LSLOPE * x; }

// D = A(16x4 f32) * B(4x16 f32) + C  -- CDNA5 V_WMMA_F32_16X16X4_F32
__device__ __forceinline__ v8f wmma4(v2f a, v2f b, v8f c) {
  return __builtin_amdgcn_wmma_f32_16x16x4_f32(false, a, false, b, (short)0, c, false, false);
}

__device__ __forceinline__ float block_reduce_sum_256(float v, float* red, int t) {
  red[t] = v;
  __syncthreads();
  #pragma unroll
  for (int off = 128; off > 0; off >>= 1) {
    if (t < off) red[t] += red[t + off];
    __syncthreads();
  }
  float s = red[0];
  __syncthreads();
  return s;
}

// ---------------- dims ----------------
enum { kPN = 32, kPE = 128, kIN = 64, kHP = 256, kHP4 = 64, kRD = 256,
       kXS = 68, kHS = 260 };   // padded LDS strides

// ===========================================================================
// Patch stage: one workgroup (8 waves) per patch graph, fully LDS-resident.
// ===========================================================================
__global__ __launch_bounds__(256) void patch_kernel(
    const float* __restrict__ feats,
    const int*   __restrict__ psrc,
    const int*   __restrict__ pdst,
    const float* __restrict__ pew,
    const float* __restrict__ Wp1,
    const float* __restrict__ Wp2,
    const float* __restrict__ Wemb,
    const float* __restrict__ g1, const float* __restrict__ b1, const float* __restrict__ a1,
    const float* __restrict__ g2, const float* __restrict__ b2, const float* __restrict__ a2,
    float* __restrict__ nf)
{
  extern __shared__ float smem[];
  float* Xs   = smem;                 // 32*68  (reused later as H2)
  float* H1s  = Xs + kPN * kXS;       // 32*260
  float* AGG  = H1s + kPN * kHS;      // 32*260
  float* AGG2 = AGG + kPN * kHS;      // 32*68
  float* rall = AGG2 + kPN * kXS;     // 384  (r0|r1|r2)
  float* embs = rall + 384;           // 256
  float* red  = embs + 256;           // 256
  float* ews  = red + 256;            // 128
  float* dnO  = ews + kPE;            // 32
  float* dnI  = dnO + kPN;            // 32
  int* srcs = (int*)(dnI + kPN);      // 128
  int* dsts = srcs + kPE;             // 128
  int* degO = dsts + kPE;             // 32
  int* degI = degO + kPN;             // 32

  const int t    = threadIdx.x;
  const int p    = blockIdx.x;
  const int lane = t & 31;
  const int wave = t >> 5;
  const int m16  = lane & 15;
  const int kh   = (lane >> 4) << 1;  // 0 | 2
  const int hi8  = (lane >> 4) << 3;  // 0 | 8

  // load x[32,64]
  const float* xg = feats + (size_t)p * (kPN * kIN);
  for (int idx = t; idx < kPN * kIN; idx += 256)
    Xs[(idx >> 6) * kXS + (idx & 63)] = xg[idx];
  if (t < kPE) {
    srcs[t] = psrc[p * kPE + t];
    dsts[t] = pdst[p * kPE + t];
    ews[t]  = pew [p * kPE + t];
  }
  if (t < kPN) { degO[t] = 0; degI[t] = 0; }
  __syncthreads();
  if (t < kPE) {
    atomicAdd(&degO[srcs[t]], 1);
    atomicAdd(&degI[dsts[t]], 1);
  }
  __syncthreads();
  if (t < kPN) {
    dnO[t] = rsqrtf(fmaxf((float)degO[t], 1.f));
    dnI[t] = rsqrtf(fmaxf((float)degI[t], 1.f));
  }
  if (t < kIN) {                       // r0 = column means of raw x
    float s = 0.f;
    for (int n = 0; n < kPN; ++n) s += Xs[n * kXS + t];
    rall[t] = s * (1.f / kPN);
  }
  __syncthreads();
  for (int idx = t; idx < kPN * kIN; idx += 256) {  // pre-scale rows by outdeg^-1/2
    int n = idx >> 6;
    Xs[n * kXS + (idx & 63)] *= dnO[n];
  }
  __syncthreads();

  // ---- conv1 GEMM: [32x64] @ Wp1[64x256] -> H1s via fp32 WMMA
  for (int rep = 0; rep < 4; ++rep) {
    const int tile = wave + rep * 8;   // 0..31
    const int mt = tile & 1;
    const int nt = tile >> 1;          // 0..15
    const float* arow = Xs + (mt * 16 + m16) * kXS + kh;
    const float* bcol = Wp1 + nt * 16 + m16;
    v8f acc = {0.f, 0.f, 0.f, 0.f, 0.f, 0.f, 0.f, 0.f};
    #pragma unroll
    for (int k0 = 0; k0 < kIN; k0 += 4) {
      v2f a, b;
      a.x = arow[k0];
      a.y = arow[k0 + 1];
      b.x = bcol[(k0 + kh) * kHP];
      b.y = bcol[(k0 + kh + 1) * kHP];
      acc = wmma4(a, b, acc);
    }
    #pragma unroll
    for (int v = 0; v < 8; ++v)
      H1s[(mt * 16 + hi8 + v) * kHS + nt * 16 + m16] = acc[v];
  }
  __syncthreads();

  // zero AGG, deterministic per-channel edge aggregation (thread t owns channel t)
  for (int idx = t; idx < kPN * kHP; idx += 256)
    AGG[(idx >> 8) * kHS + (idx & 255)] = 0.f;
  __syncthreads();
  {
    const int c = t;
    for (int e = 0; e < kPE; ++e)
      AGG[dsts[e] * kHS + c] += H1s[srcs[e] * kHS + c] * ews[e];
  }
  __syncthreads();
  // indeg^-1/2 scale + GraphNorm + lrelu; store conv2 input pre-scaled by outdeg^-1/2
  {
    const int c = t;
    const float al = a1[c], gg = g1[c], bb = b1[c];
    float s = 0.f;
    for (int n = 0; n < kPN; ++n) s += AGG[n * kHS + c] * dnI[n];
    const float mean = s * (1.f / kPN);
    float s2 = 0.f;
    for (int n = 0; n < kPN; ++n) {
      float d = AGG[n * kHS + c] * dnI[n] - al * mean;
      s2 += d * d;
    }
    const float inv = rsqrtf(s2 * (1.f / kPN) + EPS_N);
    float r = 0.f;
    for (int n = 0; n < kPN; ++n) {
      float d = AGG[n * kHS + c] * dnI[n] - al * mean;
      float o = lrelu(gg * d * inv + bb);
      r += o;
      AGG[n * kHS + c] = o * dnO[n];
    }
    rall[64 + c] = r * (1.f / kPN);
  }
  __syncthreads();

  // ---- conv2 GEMM: [32x256] @ Wp2[256x64] -> H2 (stored in Xs region)
  {
    const int mt = wave & 1;
    const int nt = wave >> 1;          // 0..3
    const float* arow = AGG + (mt * 16 + m16) * kHS + kh;
    const float* bcol = Wp2 + nt * 16 + m16;
    v8f acc = {0.f, 0.f, 0.f, 0.f, 0.f, 0.f, 0.f, 0.f};
    for (int k0 = 0; k0 < kHP; k0 += 4) {
      v2f a, b;
      a.x = arow[k0];
      a.y = arow[k0 + 1];
      b.x = bcol[(k0 + kh) * kHP4];
      b.y = bcol[(k0 + kh + 1) * kHP4];
      acc = wmma4(a, b, acc);
    }
    #pragma unroll
    for (int v = 0; v < 8; ++v)
      Xs[(mt * 16 + hi8 + v) * kXS + nt * 16 + m16] = acc[v];
  }
  __syncthreads();
  for (int idx = t; idx < kPN * kHP4; idx += 256)
    AGG2[(idx >> 6) * kXS + (idx & 63)] = 0.f;
  __syncthreads();
  if (t < kHP4) {
    const int c = t;
    for (int e = 0; e < kPE; ++e)
      AGG2[dsts[e] * kXS + c] += Xs[srcs[e] * kXS + c] * ews[e];
  }
  __syncthreads();
  if (t < kHP4) {
    const int c = t;
    const float al = a2[c], gg = g2[c], bb = b2[c];
    float s = 0.f;
    for (int n = 0; n < kPN; ++n) s += AGG2[n * kXS + c] * dnI[n];
    const float mean = s * (1.f / kPN);
    float s2 = 0.f;
    for (int n = 0; n < kPN; ++n) {
      float d = AGG2[n * kXS + c] * dnI[n] - al * mean;
      s2 += d * d;
    }
    const float inv = rsqrtf(s2 * (1.f / kPN) + EPS_N);
    float r = 0.f;
    for (int n = 0; n < kPN; ++n) {
      float d = AGG2[n * kXS + c] * dnI[n] - al * mean;
      r += lrelu(gg * d * inv + bb);
    }
    rall[320 + c] = r * (1.f / kPN);
  }
  __syncthreads();

  // readout projection: rall[384] @ Wemb[384x256]
  {
    float s = 0.f;
    for (int i = 0; i < 384; ++i) s += rall[i] * Wemb[i * kRD + t];
    embs[t] = s;
  }
  __syncthreads();
  // instance norm over the 256-dim row + lrelu
  const float mu  = block_reduce_sum_256(embs[t], red, t) * (1.f / 256.f);
  const float dd  = embs[t] - mu;
  const float var = block_reduce_sum_256(dd * dd, red, t) * (1.f / 256.f);
  nf[(size_t)p * kRD + t] = lrelu(dd * rsqrtf(var + EPS_N));
}

// ===========================================================================
// Mesh stage kernels
// ===========================================================================
__global__ __launch_bounds__(256) void mesh_deg_kernel(
    const int* __restrict__ src, const int* __restrict__ dst,
    float* __restrict__ outd, float* __restrict__ ind)
{
  const int gid = blockIdx.x * 256 + threadIdx.x;   // B*ME = 65536 exactly
  const int b = gid >> 14;                          // ME = 16384
  unsafeAtomicAdd(&outd[(b << 10) + src[gid]], 1.f);
  unsafeAtomicAdd(&ind [(b << 10) + dst[gid]], 1.f);
}

// C[b] = rowscale(A[b]) @ W ; one 16x16 tile per wave, fp32 WMMA K-steps of 4
__global__ __launch_bounds__(256) void mesh_gemm_kernel(
    const float* __restrict__ A, const float* __restrict__ W, float* __restrict__ out,
    const float* __restrict__ rowdeg, int K, int N)
{
  const int b    = blockIdx.y;
  const int t    = threadIdx.x;
  const int lane = t & 31, wave = t >> 5;
  const int m16  = lane & 15;
  const int kh   = (lane >> 4) << 1;
  const int hi8  = (lane >> 4) << 3;
  const int ntiles = N >> 4;
  const int tile = blockIdx.x * 8 + wave;
  const int mt = tile / ntiles;
  const int nt = tile - mt * ntiles;
  const int arowi = mt * 16 + m16;
  const float sc = rowdeg ? rsqrtf(fmaxf(rowdeg[(b << 10) + arowi], 1.f)) : 1.f;
  const float* arow = A + ((size_t)b * 1024 + arowi) * K + kh;
  const float* bcol = W + nt * 16 + m16;
  v8f acc = {0.f, 0.f, 0.f, 0.f, 0.f, 0.f, 0.f, 0.f};
  for (int k0 = 0; k0 < K; k0 += 4) {
    v2f a, bb;
    a.x  = arow[k0] * sc;
    a.y  = arow[k0 + 1] * sc;
    bb.x = bcol[(size_t)(k0 + kh) * N];
    bb.y = bcol[(size_t)(k0 + kh + 1) * N];
    acc = wmma4(a, bb, acc);
  }
  float* orow = out + ((size_t)b * 1024 + mt * 16 + hi8) * N + nt * 16 + m16;
  #pragma unroll
  for (int v = 0; v < 8; ++v)
    orow[(size_t)v * N] = acc[v];
}

// agg[dst] += h[src] * ew ; one thread = (edge, 8-channel chunk), C = 512
__global__ __launch_bounds__(256) void mesh_agg_kernel(
    const float* __restrict__ h, const int* __restrict__ src, const int* __restrict__ dst,
    const float* __restrict__ ew, float* __restrict__ agg)
{
  const int gid = blockIdx.x * 256 + threadIdx.x;   // B*ME*64 exactly
  const int e  = gid >> 6;                          // global edge id (b*ME+ei)
  const int c0 = (gid & 63) << 3;
  const int b  = e >> 14;
  const int s  = src[e], d = dst[e];
  const float w = ew[e];
  const float* hs = h   + ((size_t)((b << 10) + s)) * 512 + c0;
  float*       ad = agg + ((size_t)((b << 10) + d)) * 512 + c0;
  #pragma unroll
  for (int j = 0; j < 8; ++j)
    unsafeAtomicAdd(&ad[j], hs[j] * w);
}

// per-(mesh, channel): indeg scale + GraphNorm + lrelu + mean readout
// (optionally writes next-conv input pre-scaled by outdeg^-1/2)
__global__ __launch_bounds__(256) void mesh_norm_kernel(
    const float* __restrict__ agg, const float* __restrict__ indeg,
    const float* __restrict__ outdeg,
    const float* __restrict__ g, const float* __restrict__ be, const float* __restrict__ al,
    float* __restrict__ hout, float* __restrict__ r, int C)
{
  __shared__ float red[256];
  const int c = blockIdx.x, b = blockIdx.y, t = threadIdx.x;
  const float* ab  = agg + (size_t)b * 1024 * C + c;
  const float* din = indeg + (b << 10);
  const float* dou = outdeg + (b << 10);
  float s = 0.f;
  for (int i = t; i < 1024; i += 256)
    s += ab[(size_t)i * C] * rsqrtf(fmaxf(din[i], 1.f));
  const float mean = block_reduce_sum_256(s, red, t) * (1.f / 1024.f);
  const float alc = al[c];
  float s2 = 0.f;
  for (int i = t; i < 1024; i += 256) {
    float d = ab[(size_t)i * C] * rsqrtf(fmaxf(din[i], 1.f)) - alc * mean;
    s2 += d * d;
  }
  const float inv = rsqrtf(block_reduce_sum_256(s2, red, t) * (1.f / 1024.f) + EPS_N);
  const float gc = g[c], bc = be[c];
  float racc = 0.f;
  float* hb = hout ? hout + (size_t)b * 1024 * C + c : nullptr;
  for (int i = t; i < 1024; i += 256) {
    float d = ab[(size_t)i * C] * rsqrtf(fmaxf(din[i], 1.f)) - alc * mean;
    float o = lrelu(gc * d * inv + bc);
    racc += o;
    if (hb) hb[(size_t)i * C] = o * rsqrtf(fmaxf(dou[i], 1.f));
  }
  const float rsum = block_reduce_sum_256(racc, red, t);
  if (t == 0) r[b * C + c] = rsum * (1.f / 1024.f);
}

__global__ __launch_bounds__(256) void block_vec_kernel(
    const float* __restrict__ r1, const float* __restrict__ r2, float* __restrict__ bv)
{
  const int gid = blockIdx.x * 256 + threadIdx.x;   // 4096
  const int b = gid >> 10, j = gid & 1023;
  const float v = (j < 512) ? r1[(b << 9) + j] : r2[(b << 9) + j - 512];
  bv[gid] = lrelu(v);
}

__global__ __launch_bounds__(256) void final_out_kernel(
    const float* __restrict__ bv, const float* __restrict__ Wc, float* __restrict__ out)
{
  __shared__ float red[256];
  const int o = blockIdx.x, t = threadIdx.x;
  float s = 0.f;
  for (int i = t; i < 4096; i += 256) s += bv[i] * Wc[i * 16 + o];
  const float total = block_reduce_sum_256(s, red, t);
  if (t == 0) out[o] = total;
}

// ===========================================================================
extern "C" void kernel_launch(void* const* d_in, const int* in_sizes, int n_in,
                              void* d_out, int out_size, void* d_ws, size_t ws_size,
                              hipStream_t stream)
{
  const float* feats = (const float*)d_in[0];
  const int*   psrc  = (const int*)  d_in[1];
  const int*   pdst  = (const int*)  d_in[2];
  const float* pew   = (const float*)d_in[3];
  const int*   msrc  = (const int*)  d_in[4];
  const int*   mdst  = (const int*)  d_in[5];
  const float* mew   = (const float*)d_in[6];
  const float* Wp1   = (const float*)d_in[7];
  const float* Wp2   = (const float*)d_in[8];
  const float* Wemb  = (const float*)d_in[9];
  const float* gp1g  = (const float*)d_in[10];
  const float* gp1b  = (const float*)d_in[11];
  const float* gp1a  = (const float*)d_in[12];
  const float* gp2g  = (const float*)d_in[13];
  const float* gp2b  = (const float*)d_in[14];
  const float* gp2a  = (const float*)d_in[15];
  const float* gm1g  = (const float*)d_in[16];
  const float* gm1b  = (const float*)d_in[17];
  const float* gm1a  = (const float*)d_in[18];
  const float* gm2g  = (const float*)d_in[19];
  const float* gm2b  = (const float*)d_in[20];
  const float* gm2a  = (const float*)d_in[21];
  const float* Wm1   = (const float*)d_in[22];
  const float* Wm2   = (const float*)d_in[23];
  const float* Wc    = (const float*)d_in[24];

  float* ws = (float*)d_ws;
  size_t off = 0;
  float* nf   = ws + off; off += (size_t)4096 * 256;       // patch embeddings
  float* outd = ws + off; off += 4096;                     // mesh out-degrees
  float* ind  = ws + off; off += 4096;                     // mesh in-degrees
  float* bufA = ws + off; off += (size_t)4 * 1024 * 512;
  float* bufB = ws + off; off += (size_t)4 * 1024 * 512;
  float* bufC = ws + off; off += (size_t)4 * 1024 * 512;
  float* r1   = ws + off; off += 4 * 512;
  float* r2   = ws + off; off += 4 * 512;
  float* bv   = ws + off; off += 4096;

  const size_t patch_smem =
      (size_t)(kPN * kXS + kPN * kHS + kPN * kHS + kPN * kXS +
               384 + 256 + 256 + kPE + kPN + kPN) * sizeof(float) +
      (size_t)(kPE + kPE + kPN + kPN) * sizeof(int);   // ~89.6 KB of 320 KB LDS

  patch_kernel<<<4096, 256, patch_smem, stream>>>(
      feats, psrc, pdst, pew, Wp1, Wp2, Wemb,
      gp1g, gp1b, gp1a, gp2g, gp2b, gp2a, nf);

  hipMemsetAsync(outd, 0, 4096 * sizeof(float), stream);
  hipMemsetAsync(ind,  0, 4096 * sizeof(float), stream);
  mesh_deg_kernel<<<(4 * 16384) / 256, 256, 0, stream>>>(msrc, mdst, outd, ind);

  // conv1: nf (row-scaled) @ Wm1[256x512] -> bufA   (64*32 tiles / 8 waves = 256 blocks)
  mesh_gemm_kernel<<<dim3(256, 4), 256, 0, stream>>>(nf, Wm1, bufA, outd, 256, 512);
  hipMemsetAsync(bufB, 0, (size_t)4 * 1024 * 512 * sizeof(float), stream);
  mesh_agg_kernel<<<(4 * 16384 * 64) / 256, 256, 0, stream>>>(bufA, msrc, mdst, mew, bufB);
  mesh_norm_kernel<<<dim3(512, 4), 256, 0, stream>>>(
      bufB, ind, outd, gm1g, gm1b, gm1a, bufA, r1, 512);

  // conv2: bufA (already pre-scaled) @ Wm2[512x512] -> bufB
  mesh_gemm_kernel<<<dim3(256, 4), 256, 0, stream>>>(bufA, Wm2, bufB, nullptr, 512, 512);
  hipMemsetAsync(bufC, 0, (size_t)4 * 1024 * 512 * sizeof(float), stream);
  mesh_agg_kernel<<<(4 * 16384 * 64) / 256, 256, 0, stream>>>(bufB, msrc, mdst, mew, bufC);
  mesh_norm_kernel<<<dim3(512, 4), 256, 0, stream>>>(
      bufC, ind, outd, gm2g, gm2b, gm2a, nullptr, r2, 512);

  block_vec_kernel<<<16, 256, 0, stream>>>(r1, r2, bv);
  final_out_kernel<<<16, 256, 0, stream>>>(bv, Wc, (float*)d_out);

  (void)in_sizes; (void)n_in; (void)out_size; (void)ws_size;
}